// CNNRNN_63625645523427
// MI455X (gfx1250) — compile-verified
//
#include <hip/hip_runtime.h>

typedef float v2f __attribute__((ext_vector_type(2)));
typedef float v8f __attribute__((ext_vector_type(8)));

#define Bsz 128
#define Tsz 20
#define Csz 7178
#define Lsz 512
#define Fsz 512

__device__ __forceinline__ v8f wmma_f32_k4(v2f a, v2f b, v8f c) {
    // V_WMMA_F32_16X16X4_F32: D = A(16x4) x B(4x16) + C(16x16), full f32
    return __builtin_amdgcn_wmma_f32_16x16x4_f32(
        /*neg_a=*/false, a, /*neg_b=*/false, b,
        /*c_mod=*/(short)0, c, /*reuse_a=*/false, /*reuse_b=*/false);
}

// Generic NT GEMM: out[m,n] = act( sum_k A[m,k]*Wt[n,k] + bias[n] + add[m,n] )
// Wave tile: (MT*16)(M) x 64(N); block = 4 waves along N -> (MT*16) x 256 tile.
// Grid: (M/(16*MT), ceil(N/256)). All lanes stay live (WMMA needs EXEC all-1s).
// Interior blocks take a mask-free fast path (uniform branch); only the one
// N-edge column-block pays the cndmask cost.
template <int MT>
__global__ __launch_bounds__(128) void gemm_nt_wmma(
    const float* __restrict__ A, int lda,
    const float* __restrict__ Wt, int ldw,
    const float* __restrict__ bias,
    const float* __restrict__ add, int ldadd,
    float* __restrict__ out, int ldc,
    int N, int K, int doRelu)
{
    const int wave = threadIdx.x >> 5;
    const int lane = threadIdx.x & 31;
    const int half = lane >> 4;   // lanes 16-31 hold K+2 / M+8 halves
    const int lm   = lane & 15;
    const int m0 = blockIdx.x * (16 * MT);
    const int n0 = blockIdx.y * 256 + wave * 64;

    const float* pA[MT];
#pragma unroll
    for (int t = 0; t < MT; ++t)
        pA[t] = A + (size_t)(m0 + t * 16 + lm) * lda + 2 * half;

    const float* pB[4];
    bool vB[4];
#pragma unroll
    for (int j = 0; j < 4; ++j) {
        int n = n0 + j * 16 + lm;
        vB[j] = (n < N);
        pB[j] = Wt + (size_t)(vB[j] ? n : 0) * ldw + 2 * half;
    }

    v8f acc[MT][4] = {};

    if (n0 + 63 < N) {
        // interior fast path: no masking at all
        for (int k = 0; k < K; k += 4) {
            v2f a[MT];
#pragma unroll
            for (int t = 0; t < MT; ++t) a[t] = *(const v2f*)(pA[t] + k);
#pragma unroll
            for (int j = 0; j < 4; ++j) {
                v2f b = *(const v2f*)(pB[j] + k);
#pragma unroll
                for (int t = 0; t < MT; ++t)
                    acc[t][j] = wmma_f32_k4(a[t], b, acc[t][j]);
            }
        }
    } else {
        // N-edge path: zeroed fragment loads keep EXEC all-1s for WMMA
        for (int k = 0; k < K; k += 4) {
            v2f a[MT];
#pragma unroll
            for (int t = 0; t < MT; ++t) a[t] = *(const v2f*)(pA[t] + k);
#pragma unroll
            for (int j = 0; j < 4; ++j) {
                v2f b = *(const v2f*)(pB[j] + k);
                if (!vB[j]) { b.x = 0.0f; b.y = 0.0f; }
#pragma unroll
                for (int t = 0; t < MT; ++t)
                    acc[t][j] = wmma_f32_k4(a[t], b, acc[t][j]);
            }
        }
    }

#pragma unroll
    for (int j = 0; j < 4; ++j) {
        int n = n0 + j * 16 + lm;
        if (n < N) {
            float bv = bias ? bias[n] : 0.0f;
#pragma unroll
            for (int t = 0; t < MT; ++t) {
#pragma unroll
                for (int r = 0; r < 8; ++r) {
                    int m = m0 + t * 16 + r + 8 * half;  // C/D: VGPR r -> M=r (+8 hi lanes)
                    float v = acc[t][j][r] + bv;
                    if (add) v += add[(size_t)m * ldadd + n];
                    if (doRelu) v = fmaxf(v, 0.0f);
                    out[(size_t)m * ldc + n] = v;
                }
            }
        }
    }
}

// gates = U_l[labels[:,s]] @ W_ih^T + h @ W_hh^T + b_ih + b_hh   (128 x 2048)
// Embedding gather folded into the A-fragment row pointer. N exact -> no masks.
__global__ __launch_bounds__(128) void gates_wmma(
    const float* __restrict__ Ul,
    const int*   __restrict__ labels, int step,
    const float* __restrict__ h,
    const float* __restrict__ Wih,
    const float* __restrict__ Whh,
    const float* __restrict__ bih,
    const float* __restrict__ bhh,
    float* __restrict__ gates)
{
    const int wave = threadIdx.x >> 5;
    const int lane = threadIdx.x & 31;
    const int half = lane >> 4;
    const int lm   = lane & 15;
    const int m0 = blockIdx.x << 4;
    const int n0 = blockIdx.y * 256 + wave * 64;
    const int row = m0 + lm;

    const float* pA1 = Ul + (size_t)labels[row * Tsz + step] * Lsz + 2 * half;
    const float* pA2 = h  + (size_t)row * Fsz + 2 * half;

    const float* pB1[4];
    const float* pB2[4];
#pragma unroll
    for (int j = 0; j < 4; ++j) {
        int n = n0 + j * 16 + lm;
        pB1[j] = Wih + (size_t)n * Lsz + 2 * half;
        pB2[j] = Whh + (size_t)n * Fsz + 2 * half;
    }

    v8f acc[4] = {};
    for (int k = 0; k < Lsz; k += 4) {
        v2f a = *(const v2f*)(pA1 + k);
#pragma unroll
        for (int j = 0; j < 4; ++j)
            acc[j] = wmma_f32_k4(a, *(const v2f*)(pB1[j] + k), acc[j]);
    }
    for (int k = 0; k < Fsz; k += 4) {
        v2f a = *(const v2f*)(pA2 + k);
#pragma unroll
        for (int j = 0; j < 4; ++j)
            acc[j] = wmma_f32_k4(a, *(const v2f*)(pB2[j] + k), acc[j]);
    }

#pragma unroll
    for (int j = 0; j < 4; ++j) {
        int n = n0 + j * 16 + lm;
        float bv = bih[n] + bhh[n];
#pragma unroll
        for (int r = 0; r < 8; ++r) {
            int m = m0 + r + 8 * half;
            gates[(size_t)m * (4 * Fsz) + n] = acc[j][r] + bv;
        }
    }
}

__global__ void init_hc(const float* __restrict__ h0, const float* __restrict__ c0,
                        float* __restrict__ h, float* __restrict__ c) {
    int i = blockIdx.x * blockDim.x + threadIdx.x;   // 128*512 threads
    h[i] = h0[i & (Fsz - 1)];
    c[i] = c0[i & (Fsz - 1)];
}

__global__ void write_x0(const float* __restrict__ P, float* __restrict__ X) {
    int i = blockIdx.x * blockDim.x + threadIdx.x;   // 128*512 threads
    int b = i >> 9, j = i & 511;
    X[(size_t)b * Tsz * Lsz + j] = fmaxf(P[i], 0.0f);
}

__global__ void combine_gates(const float* __restrict__ gates,
                              float* __restrict__ h, float* __restrict__ c) {
    int i = blockIdx.x * blockDim.x + threadIdx.x;   // 128*512 threads
    int m = i >> 9, j = i & 511;
    const float* g = gates + (size_t)m * (4 * Fsz);
    float ig = fmaxf(g[j],           0.0f);
    float fg = fmaxf(g[Fsz + j],     0.0f);
    float cg = fmaxf(g[2 * Fsz + j], 0.0f);
    float og = fmaxf(g[3 * Fsz + j], 0.0f);
    float nc = fg * c[i] + ig * cg;
    float nh = og * fmaxf(nc, 0.0f);
    c[i] = nc;
    h[i] = nh;
}

__global__ void transpose512(const float* __restrict__ src, float* __restrict__ dst) {
    int x = blockIdx.x * 16 + threadIdx.x;
    int y = blockIdx.y * 16 + threadIdx.y;
    dst[(size_t)x * 512 + y] = src[(size_t)y * 512 + x];
}

extern "C" void kernel_launch(void* const* d_in, const int* in_sizes, int n_in,
                              void* d_out, int out_size, void* d_ws, size_t ws_size,
                              hipStream_t stream) {
    const float* img    = (const float*)d_in[0];
    const int*   labels = (const int*)  d_in[1];
    const float* Ul     = (const float*)d_in[2];
    const float* projI  = (const float*)d_in[3];
    const float* projO  = (const float*)d_in[4];
    const float* Wih    = (const float*)d_in[5];
    const float* bih    = (const float*)d_in[6];
    const float* Whh    = (const float*)d_in[7];
    const float* bhh    = (const float*)d_in[8];
    const float* h0     = (const float*)d_in[9];
    const float* c0     = (const float*)d_in[10];
    float* out = (float*)d_out;

    // workspace layout (floats), ~8.8 MB total
    float* ws    = (float*)d_ws;
    float* P     = ws;                        // 128*512   pre-relu proj_img
    float* h     = P     + Bsz * Fsz;         // 128*512
    float* c     = h     + Bsz * Fsz;         // 128*512
    float* gates = c     + Bsz * Fsz;         // 128*2048
    float* PIt   = gates + Bsz * 4 * Fsz;     // 512*512   proj_I^T
    float* POt   = PIt   + Fsz * Lsz;         // 512*512   proj_O^T
    float* X     = POt   + Fsz * Lsz;         // 2560*512  all x_t, row = b*T + t

    dim3 tb16(16, 16);
    transpose512<<<dim3(32, 32), tb16, 0, stream>>>(projI, PIt);
    transpose512<<<dim3(32, 32), tb16, 0, stream>>>(projO, POt);
    init_hc<<<(Bsz * Fsz) / 256, 256, 0, stream>>>(h0, c0, h, c);

    // P = img @ proj_I  (no activation; x0 = relu(P) written separately)
    gemm_nt_wmma<1><<<dim3(Bsz / 16, (Lsz + 255) / 256), 128, 0, stream>>>(
        img, Fsz, PIt, Fsz, nullptr, nullptr, 0, P, Lsz, Lsz, Fsz, 0);
    write_x0<<<(Bsz * Fsz) / 256, 256, 0, stream>>>(P, X);

    for (int s = 0; s < Tsz - 1; ++s) {
        gates_wmma<<<dim3(Bsz / 16, (4 * Fsz) / 256), 128, 0, stream>>>(
            Ul, labels, s, h, Wih, Whh, bih, bhh, gates);
        combine_gates<<<(Bsz * Fsz) / 256, 256, 0, stream>>>(gates, h, c);
        // x_{s+1} = relu(P + nh @ proj_O) -> X rows (b*T + s + 1)
        gemm_nt_wmma<1><<<dim3(Bsz / 16, (Lsz + 255) / 256), 128, 0, stream>>>(
            h, Fsz, POt, Fsz, nullptr, P, Lsz,
            X + (size_t)(s + 1) * Lsz, Tsz * Lsz, Lsz, Fsz, 1);
    }

    // Final fused vocab GEMM: out(2560 x 7178) = X(2560 x 512) @ U_l^T
    // MT=2 -> 32x64 wave tile, 10.7 FLOP/B from L2; grid 80 x 29.
    gemm_nt_wmma<2><<<dim3((Bsz * Tsz) / 32, (Csz + 255) / 256), 128, 0, stream>>>(
        X, Lsz, Ul, Lsz, nullptr, nullptr, 0, out, Csz, Csz, Lsz, 0);
}